// LoopModel_26465588478162
// MI455X (gfx1250) — compile-verified
//
#include <hip/hip_runtime.h>
#include <math.h>

typedef __attribute__((ext_vector_type(2))) float v2f;
typedef __attribute__((ext_vector_type(8))) float v8f;

#define E_N    256
#define C_N    128
#define HW_N   3136   // 56*56
#define NC_N   200
#define NP_N   512
#define NL_N   64
#define LL_N   8
#define EDGE_DIM 256  // 2*C
#define LOOP_IN  258
#define K0_PAD 784    // 3*258=774 padded to multiple of 16

// ---------------------------------------------------------------------------
// 1) Bandwidth-bound: per-(e,c) spatial max + concat coord  -> edge_x [256,256]
// ---------------------------------------------------------------------------
__global__ void __launch_bounds__(256) lm_maxpool_concat(
    const float* __restrict__ image_x, const float* __restrict__ coord_x,
    float* __restrict__ edge_x)
{
  int wave = threadIdx.x >> 5, lane = threadIdx.x & 31;
  int row  = blockIdx.x * 8 + wave;              // [0, 32768)
  const float4* p = (const float4*)(image_x + (size_t)row * HW_N);
  float m = -3.402823466e38f;
  for (int i = lane; i < 784; i += 32) {
    float4 v = p[i];
    if (i + 128 < 784) __builtin_prefetch((const void*)(p + i + 128), 0, 0);
    m = fmaxf(m, fmaxf(fmaxf(v.x, v.y), fmaxf(v.z, v.w)));
  }
  for (int off = 16; off > 0; off >>= 1)
    m = fmaxf(m, __shfl_xor(m, off, 32));
  if (lane == 0) {
    int e = row >> 7, c = row & 127;
    edge_x[e * EDGE_DIM + c]       = m;
    edge_x[e * EDGE_DIM + 128 + c] = coord_x[row];
  }
}

// ---------------------------------------------------------------------------
// 2) Weight prep -> pair-interleaved K-major:  Btp[(k>>1)*2N + 2n + (k&1)]
//    holds W^T[k][n]; rows k >= Kr are zero (padding). Conv weights use
//    k = dk*I + ci  ->  W[n*Kr + ci*3 + dk].
// ---------------------------------------------------------------------------
__global__ void lm_prep_bt(const float* __restrict__ W, float* __restrict__ Btp,
                           int N, int Kr, int Kpad, int I)
{
  int idx = blockIdx.x * blockDim.x + threadIdx.x;
  if (idx >= N * Kpad) return;
  int k = idx / N, n = idx % N;
  float v = 0.f;
  if (k < Kr) {
    if (I > 0) { int dk = k / I, ci = k % I; v = W[n * Kr + ci * 3 + dk]; }
    else       { v = W[n * Kr + k]; }
  }
  Btp[(k >> 1) * (2 * N) + 2 * n + (k & 1)] = v;
}

// ---------------------------------------------------------------------------
// 3) fp32 WMMA GEMM: C[M,N] = act(A[M,K] @ B + bias).  K % 16 == 0,
//    M,N % 16 == 0 (padded by construction). One wave per 16x16 tile.
//    8x global_load_b64 + 4x v_wmma_f32_16x16x4_f32 per K=16 step,
//    no predication -> EXEC stays all-1s, loads pipeline freely.
// ---------------------------------------------------------------------------
__global__ void __launch_bounds__(32) lm_wmma_gemm_f32(
    const float* __restrict__ A, const float* __restrict__ Btp,
    const float* __restrict__ bias, float* __restrict__ C,
    int K, int N, int act, int ntiles_n)
{
  int tile = blockIdx.x;
  int m0 = (tile / ntiles_n) * 16;
  int n0 = (tile % ntiles_n) * 16;
  int lane = threadIdx.x & 31;
  int half = lane >> 4;                // 0 -> K pair (k,k+1); 1 -> (k+2,k+3)
  int r    = lane & 15;
  int bcol = n0 + r;
  const v2f* Ar = (const v2f*)(A + (size_t)(m0 + r) * K);
  const v2f* Bp = (const v2f*)Btp;
  v8f acc = {};
  for (int k2 = 0; k2 < (K >> 1); k2 += 8) {     // k2 = k/2, 16 K per iter
    int p = k2 + half;
    v2f a0 = Ar[p + 0];
    v2f a1 = Ar[p + 2];
    v2f a2 = Ar[p + 4];
    v2f a3 = Ar[p + 6];
    v2f b0 = Bp[(size_t)(p + 0) * N + bcol];
    v2f b1 = Bp[(size_t)(p + 2) * N + bcol];
    v2f b2 = Bp[(size_t)(p + 4) * N + bcol];
    v2f b3 = Bp[(size_t)(p + 6) * N + bcol];
    acc = __builtin_amdgcn_wmma_f32_16x16x4_f32(false, a0, false, b0, (short)0, acc, false, false);
    acc = __builtin_amdgcn_wmma_f32_16x16x4_f32(false, a1, false, b1, (short)0, acc, false, false);
    acc = __builtin_amdgcn_wmma_f32_16x16x4_f32(false, a2, false, b2, (short)0, acc, false, false);
    acc = __builtin_amdgcn_wmma_f32_16x16x4_f32(false, a3, false, b3, (short)0, acc, false, false);
  }
  float bv = bias ? bias[bcol] : 0.f;
  for (int j = 0; j < 8; ++j) {
    float v = acc[j] + bv;
    if (act) v = fmaxf(v, 0.f);
    C[(size_t)(m0 + j + 8 * half) * N + bcol] = v;
  }
}

// ---------------------------------------------------------------------------
// 4) edge_conf = sigmoid(h1[e,:64] . W_ep2 + b_ep2), one wave per edge
// ---------------------------------------------------------------------------
__global__ void __launch_bounds__(32) lm_edge_conf(
    const float* __restrict__ h1, const float* __restrict__ W2,
    const float* __restrict__ b2, float* __restrict__ conf)
{
  int e = blockIdx.x, lane = threadIdx.x;
  float s = h1[e * 64 + lane] * W2[lane] + h1[e * 64 + 32 + lane] * W2[32 + lane];
  for (int off = 16; off > 0; off >>= 1) s += __shfl_xor(s, off, 32);
  if (lane == 0) conf[e] = 1.f / (1.f + expf(-(s + b2[0])));
}

// ---------------------------------------------------------------------------
// 5) Deterministic scatter-mean: per-corner scan of all 512 pairs
// ---------------------------------------------------------------------------
__global__ void __launch_bounds__(256) lm_corner_mean(
    const float* __restrict__ edge_x, const int* __restrict__ pairs,
    float* __restrict__ cf)
{
  int c = blockIdx.x, t = threadIdx.x;
  float acc = 0.f; int cnt = 0;
  for (int p = 0; p < NP_N; ++p) {
    if (pairs[2 * p] == c) { acc += edge_x[pairs[2 * p + 1] * EDGE_DIM + t]; cnt++; }
  }
  cf[c * EDGE_DIM + t] = acc / fmaxf((float)cnt, 1.f);
}

// ---------------------------------------------------------------------------
// 6) loop_inp x0[row=nl*8+t, ch]
// ---------------------------------------------------------------------------
__global__ void lm_build_x0(const float* __restrict__ cf,
                            const float* __restrict__ corners,
                            const int* __restrict__ loops, float* __restrict__ x0)
{
  int idx = blockIdx.x * blockDim.x + threadIdx.x;
  if (idx >= 512 * LOOP_IN) return;
  int row = idx / LOOP_IN, ch = idx % LOOP_IN;
  int corner = loops[row];
  x0[idx] = (ch < 256) ? cf[corner * EDGE_DIM + ch] : corners[corner * 2 + (ch - 256)];
}

// ---------------------------------------------------------------------------
// 7) im2col with ReflectionPad1d(1); zero-fills pad columns kk in [3*Cin,Kpad)
// ---------------------------------------------------------------------------
__global__ void lm_im2col(const float* __restrict__ src, float* __restrict__ dst,
                          int Cin, int Kpad)
{
  int Kr = 3 * Cin;
  int idx = blockIdx.x * blockDim.x + threadIdx.x;
  if (idx >= 512 * Kpad) return;
  int row = idx / Kpad, kk = idx % Kpad;
  float v = 0.f;
  if (kk < Kr) {
    int dk = kk / Cin, ci = kk % Cin;
    int nl = row >> 3, t = row & 7;
    int pos = t + dk - 1;
    pos = (pos < 0) ? 1 : ((pos > 7) ? 6 : pos);
    v = src[(nl * 8 + pos) * Cin + ci];
  }
  dst[idx] = v;
}

// ---------------------------------------------------------------------------
// 8) loop_features[nl,co] = max_t h2[(nl*8+t),co]
// ---------------------------------------------------------------------------
__global__ void lm_loop_max(const float* __restrict__ h, float* __restrict__ lf)
{
  int idx = blockIdx.x * blockDim.x + threadIdx.x;
  if (idx >= NL_N * 64) return;
  int nl = idx >> 6, co = idx & 63;
  float m = h[(nl * 8) * 64 + co];
  for (int t = 1; t < 8; ++t) m = fmaxf(m, h[(nl * 8 + t) * 64 + co]);
  lf[idx] = m;
}

// ---------------------------------------------------------------------------
// 9) loop_conf = sigmoid(lh[l,:32] . Wl2 + bl2), one wave per loop
// ---------------------------------------------------------------------------
__global__ void __launch_bounds__(32) lm_loop_conf(
    const float* __restrict__ lh, const float* __restrict__ W2,
    const float* __restrict__ b2, float* __restrict__ conf)
{
  int l = blockIdx.x, lane = threadIdx.x;
  float s = lh[l * 32 + lane] * W2[lane];
  for (int off = 16; off > 0; off >>= 1) s += __shfl_xor(s, off, 32);
  if (lane == 0) conf[l] = 1.f / (1.f + expf(-(s + b2[0])));
}

// ---------------------------------------------------------------------------
// 10) loop_edges[nl,e]
// ---------------------------------------------------------------------------
__global__ void __launch_bounds__(256) lm_loop_edges(
    const int* __restrict__ loops, const int* __restrict__ ec, float* __restrict__ le)
{
  __shared__ int lp[8];
  int nl = blockIdx.x, e = threadIdx.x;
  if (e < 8) lp[e] = loops[nl * 8 + e];
  __syncthreads();
  int a = ec[2 * e], b = ec[2 * e + 1];
  bool hit = false;
  for (int j = 0; j < 8; ++j) {
    int cur = lp[j], prv = lp[(j + 7) & 7], nxt = lp[(j + 1) & 7];
    hit = hit || (a == cur && (b == prv || b == nxt));
  }
  le[nl * E_N + e] = hit ? 1.f : 0.f;
}

// ---------------------------------------------------------------------------
// 11) coord_features[e,f]
// ---------------------------------------------------------------------------
__global__ void __launch_bounds__(64) lm_coord_feat(
    const float* __restrict__ conf, const float* __restrict__ le,
    const float* __restrict__ lf, float* __restrict__ cfeat)
{
  int e = blockIdx.x, f = threadIdx.x;
  float num = 0.f, den = 0.f;
  for (int nl = 0; nl < NL_N; ++nl) {
    float w = conf[nl] * le[nl * E_N + e];
    den += w;
    num += w * lf[nl * 64 + f];
  }
  cfeat[e * 64 + f] = num / fmaxf(den, 1e-4f);
}

// ---------------------------------------------------------------------------
// 12) A_final[e, 0:128]=coord_x, [128:192]=coord_features
// ---------------------------------------------------------------------------
__global__ void lm_build_afin(const float* __restrict__ coord_x,
                              const float* __restrict__ cfeat, float* __restrict__ afin)
{
  int idx = blockIdx.x * blockDim.x + threadIdx.x;
  if (idx >= 256 * 192) return;
  int row = idx / 192, k = idx % 192;
  afin[idx] = (k < 128) ? coord_x[row * 128 + k] : cfeat[row * 64 + (k - 128)];
}

// ===========================================================================
extern "C" void kernel_launch(void* const* d_in, const int* in_sizes, int n_in,
                              void* d_out, int out_size, void* d_ws, size_t ws_size,
                              hipStream_t stream)
{
  const float* image_x = (const float*)d_in[0];
  const float* coord_x = (const float*)d_in[1];
  const float* corners = (const float*)d_in[2];
  const int*   cep     = (const int*)d_in[3];
  const int*   ecor    = (const int*)d_in[4];
  const int*   loops   = (const int*)d_in[5];
  const float* W_ep1 = (const float*)d_in[6];  const float* b_ep1 = (const float*)d_in[7];
  const float* W_ep2 = (const float*)d_in[8];  const float* b_ep2 = (const float*)d_in[9];
  const float* Wc0 = (const float*)d_in[10];   const float* bc0 = (const float*)d_in[11];
  const float* Wc1 = (const float*)d_in[12];   const float* bc1 = (const float*)d_in[13];
  const float* Wc2 = (const float*)d_in[14];   const float* bc2 = (const float*)d_in[15];
  const float* Wl1 = (const float*)d_in[16];   const float* bl1 = (const float*)d_in[17];
  const float* Wl2 = (const float*)d_in[18];   const float* bl2 = (const float*)d_in[19];
  const float* W_agg = (const float*)d_in[20];

  float* out       = (float*)d_out;           // [256*128]
  float* edge_conf = out + 32768;             // [256]
  float* loop_conf = out + 33024;             // [64]
  float* loop_edg  = out + 33088;             // [64*256]

  float* ws = (float*)d_ws;
  float* edge_x = ws;                   // 65536
  float* h1     = edge_x + 65536;       // 16384
  float* cf     = h1     + 16384;       // 51200
  float* x0     = cf     + 51200;       // 132096  (512 x 258)
  float* col0   = x0     + 132096;      // 401408  (512 x 784)
  float* h0     = col0   + 401408;      // 32768
  float* col1   = h0     + 32768;       // 98304   (512 x 192)
  float* h1c    = col1   + 98304;       // 32768
  float* col2   = h1c    + 32768;       // 98304
  float* h2c    = col2   + 98304;       // 32768
  float* lf     = h2c    + 32768;       // 4096
  float* lh     = lf     + 4096;        // 2048
  float* cfeat  = lh     + 2048;        // 16384
  float* afin   = cfeat  + 16384;       // 49152   (256 x 192)
  float* bt_ep1 = afin   + 49152;       // 16384   (256 x 64)
  float* bt_c0  = bt_ep1 + 16384;       // 50176   (784 x 64)
  float* bt_c1  = bt_c0  + 50176;       // 12288   (192 x 64)
  float* bt_c2  = bt_c1  + 12288;       // 12288
  float* bt_l1  = bt_c2  + 12288;       // 2048    (64 x 32)
  float* bt_agg = bt_l1  + 2048;        // 24576   (192 x 128)

  // --- stage 1: HBM-bound max-pool + concat (the real work) ---
  lm_maxpool_concat<<<4096, 256, 0, stream>>>(image_x, coord_x, edge_x);

  // --- weight prep: pair-interleaved K-major, zero-padded ---
  lm_prep_bt<<<(64 * 256 + 255) / 256, 256, 0, stream>>>(W_ep1, bt_ep1, 64, 256, 256, 0);
  lm_prep_bt<<<(64 * K0_PAD + 255) / 256, 256, 0, stream>>>(Wc0, bt_c0, 64, 774, K0_PAD, 258);
  lm_prep_bt<<<(64 * 192 + 255) / 256, 256, 0, stream>>>(Wc1, bt_c1, 64, 192, 192, 64);
  lm_prep_bt<<<(64 * 192 + 255) / 256, 256, 0, stream>>>(Wc2, bt_c2, 64, 192, 192, 64);
  lm_prep_bt<<<(32 * 64 + 255) / 256, 256, 0, stream>>>(Wl1, bt_l1, 32, 64, 64, 0);
  lm_prep_bt<<<(128 * 192 + 255) / 256, 256, 0, stream>>>(W_agg, bt_agg, 128, 192, 192, 0);

  // --- edge MLP: h1 = relu(edge_x @ W_ep1^T + b), WMMA M=256 N=64 K=256 ---
  lm_wmma_gemm_f32<<<16 * 4, 32, 0, stream>>>(edge_x, bt_ep1, b_ep1, h1, 256, 64, 1, 4);
  lm_edge_conf<<<256, 32, 0, stream>>>(h1, W_ep2, b_ep2, edge_conf);

  // --- corner scatter-mean (deterministic scan) ---
  lm_corner_mean<<<NC_N, 256, 0, stream>>>(edge_x, cep, cf);

  // --- loop encoder: im2col + WMMA GEMM x3 ---
  lm_build_x0<<<(512 * 258 + 255) / 256, 256, 0, stream>>>(cf, corners, loops, x0);
  lm_im2col<<<(512 * K0_PAD + 255) / 256, 256, 0, stream>>>(x0, col0, 258, K0_PAD);
  lm_wmma_gemm_f32<<<32 * 4, 32, 0, stream>>>(col0, bt_c0, bc0, h0, K0_PAD, 64, 1, 4);
  lm_im2col<<<(512 * 192 + 255) / 256, 256, 0, stream>>>(h0, col1, 64, 192);
  lm_wmma_gemm_f32<<<32 * 4, 32, 0, stream>>>(col1, bt_c1, bc1, h1c, 192, 64, 1, 4);
  lm_im2col<<<(512 * 192 + 255) / 256, 256, 0, stream>>>(h1c, col2, 64, 192);
  lm_wmma_gemm_f32<<<32 * 4, 32, 0, stream>>>(col2, bt_c2, bc2, h2c, 192, 64, 1, 4);
  lm_loop_max<<<(NL_N * 64 + 255) / 256, 256, 0, stream>>>(h2c, lf);

  // --- loop MLP: WMMA M=64 N=32 K=64, then conf head ---
  lm_wmma_gemm_f32<<<4 * 2, 32, 0, stream>>>(lf, bt_l1, bl1, lh, 64, 32, 1, 2);
  lm_loop_conf<<<NL_N, 32, 0, stream>>>(lh, Wl2, bl2, loop_conf);

  // --- loop-edge incidence + confidence-weighted average ---
  lm_loop_edges<<<NL_N, 256, 0, stream>>>(loops, ecor, loop_edg);
  lm_coord_feat<<<E_N, 64, 0, stream>>>(loop_conf, loop_edg, lf, cfeat);

  // --- final aggregate: out = relu(afin @ W_agg^T), WMMA M=256 N=128 K=192 ---
  lm_build_afin<<<(256 * 192 + 255) / 256, 256, 0, stream>>>(coord_x, cfeat, afin);
  lm_wmma_gemm_f32<<<16 * 8, 32, 0, stream>>>(afin, bt_agg, nullptr, out, 192, 128, 1, 8);
}